// DetectPostProcess_19722489823451
// MI455X (gfx1250) — compile-verified
//
#include <hip/hip_runtime.h>
#include <stdint.h>

// Problem constants (from reference)
#define B_     32
#define N_     24564
#define C_     21
#define NCLS   20          // classes excluding background
#define K_     200
#define TH_IOU_  0.5f
#define TH_CONF_ 0.5f

// ---------------------------------------------------------------------------
// Kernel 1: fused softmax + SSD box decode.
// Memory-bound: reads conf (66MB) + loc/anchors (13MB), writes transposed
// score planes [B,20,N] (63MB) + corner boxes [B,N,4] (13MB). All stores
// coalesced. ~10us at 23.3 TB/s.
// ---------------------------------------------------------------------------
__global__ __launch_bounds__(256) void k_softmax_decode(
    const float* __restrict__ conf, const float* __restrict__ loc,
    const float* __restrict__ anchors, float* __restrict__ scoresT,
    float* __restrict__ boxes)
{
  int n = blockIdx.x * 256 + threadIdx.x;
  int b = blockIdx.y;
  if (n >= N_) return;

  size_t base = ((size_t)b * N_ + n) * C_;
  float x[C_];
  float m = -1e30f;
#pragma unroll
  for (int c = 0; c < C_; ++c) { x[c] = conf[base + c]; m = fmaxf(m, x[c]); }
  float s = 0.f;
#pragma unroll
  for (int c = 0; c < C_; ++c) { x[c] = __expf(x[c] - m); s += x[c]; }
  float inv = 1.f / s;
  // transposed per-class planes (skip background class 0): coalesced per plane
#pragma unroll
  for (int c = 1; c < C_; ++c)
    scoresT[((size_t)b * NCLS + (c - 1)) * N_ + n] = x[c] * inv;

  // decode: (cx,cy,w,h) anchors + deltas -> corners
  float4 l = reinterpret_cast<const float4*>(loc)[(size_t)b * N_ + n];
  float4 a = reinterpret_cast<const float4*>(anchors)[n];
  float cx = a.x + l.x * 0.1f * a.z;
  float cy = a.y + l.y * 0.1f * a.w;
  float w  = a.z * __expf(l.z * 0.2f);
  float h  = a.w * __expf(l.w * 0.2f);
  float4 bx = make_float4(cx - 0.5f * w, cy - 0.5f * h,
                          cx + 0.5f * w, cy + 0.5f * h);
  reinterpret_cast<float4*>(boxes)[(size_t)b * N_ + n] = bx;
}

// ---------------------------------------------------------------------------
// Kernel 2: per-(batch,class) top-K + NMS, entirely in CDNA5's 320KB LDS.
// LDS layout (dynamic, ~110.5KB -> 2 blocks/WGP):
//   [0      , 98304) : skey[24576]  score plane as sortable uint bits (TDM dst)
//   [98304  ,106496) : cand[1024]   64-bit (key desc | ~idx) composites
//   [106496 ,109696) : tbox[200*4]  top-K boxes (async-load dst)
//   [109696 ,110496) : tsc[200]     top-K scores
//   [110496 ,110504) : cnt[2]       counters
// ---------------------------------------------------------------------------
#define SKEY_OFF 0u
#define CAND_OFF 98304u
#define TBOX_OFF 106496u
#define TSC_OFF  109696u
#define CNT_OFF  110496u
#define SMEM_SZ  110592u
#define CAP      1024

typedef unsigned int u32x4 __attribute__((ext_vector_type(4)));
typedef int          i32x8 __attribute__((ext_vector_type(8)));
typedef int          i32x4 __attribute__((ext_vector_type(4)));

__global__ __launch_bounds__(256) void k_topk_nms(
    const float* __restrict__ scoresT, const float* __restrict__ boxes,
    float* __restrict__ out)
{
  extern __shared__ char smem[];
  uint32_t*           skey = (uint32_t*)(smem + SKEY_OFF);
  unsigned long long* cand = (unsigned long long*)(smem + CAND_OFF);
  float*              tbox = (float*)(smem + TBOX_OFF);
  float*              tsc  = (float*)(smem + TSC_OFF);
  int*                cnt  = (int*)(smem + CNT_OFF);

  const int tid   = threadIdx.x;
  const int plane = blockIdx.x;            // = b*20 + c
  const int b     = plane / NCLS;
  // low 32 bits of the flat LDS address == wave-relative LDS byte offset
  const uint32_t lds_base = (uint32_t)(uintptr_t)(void*)smem;

  // ---- Stage 1: Tensor Data Mover pulls the whole 98KB score plane to LDS.
  // TDM ignores EXEC, so gate at *wave* granularity (only wave 0 issues).
  const float* src = scoresT + (size_t)plane * N_;
#if __has_builtin(__builtin_amdgcn_tensor_load_to_lds)
  if (tid < 32) {
    uint64_t ga = (uint64_t)(uintptr_t)src;
    u32x4 g0;
    g0[0] = 1u;                                   // count=1, user descriptor
    g0[1] = lds_base + SKEY_OFF;                  // lds_addr
    g0[2] = (uint32_t)ga;                         // global_addr[31:0]
    g0[3] = (uint32_t)(ga >> 32) | 0x80000000u;   // global_addr[56:32] | type=2
    i32x8 g1;
    g1[0] = (int)(2u << 16);                      // data_size = 4B
    g1[1] = (int)((N_ & 0xFFFF) << 16);           // tensor_dim0[15:0]
    g1[2] = (int)((N_ >> 16) | (1 << 16));        // tensor_dim0[31:16], tensor_dim1=1
    g1[3] = (int)((N_ & 0xFFFF) << 16);           // tile_dim0 = N (fits 16 bits)
    g1[4] = 1;                                    // tile_dim1 = 1, tile_dim2 = 0
    g1[5] = N_;                                   // tensor_dim0_stride
    g1[6] = 0; g1[7] = 0;                         // tensor_dim1_stride = 0
    i32x4 z4 = {0, 0, 0, 0};
    i32x8 z8 = {0, 0, 0, 0, 0, 0, 0, 0};
    // 6-arg form (clang-23 / therock-10.0 headers): (g0, g1, g2, g3, g4, cpol)
    __builtin_amdgcn_tensor_load_to_lds(g0, g1, z4, z4, z8, 0);
#if __has_builtin(__builtin_amdgcn_s_wait_tensorcnt)
    __builtin_amdgcn_s_wait_tensorcnt(0);
#else
    asm volatile("s_wait_tensorcnt 0x0" ::: "memory");
#endif
  }
#else
  for (int j = tid; j < N_; j += 256)
    skey[j] = reinterpret_cast<const uint32_t*>(src)[j];
#endif
  __syncthreads();

  // ---- Stage 2: exact top-K threshold via bitwise binary search on float
  // bits (softmax scores are positive => uint bit order == float order).
  // Find largest T with count(key >= T) >= K.
  uint32_t T = 0;
  for (int bit = 30; bit >= 0; --bit) {           // sign bit always 0
    uint32_t Tc = T | (1u << bit);
    if (tid == 0) cnt[0] = 0;
    __syncthreads();
    int local = 0;
    for (int j = tid; j < N_; j += 256) local += (skey[j] >= Tc) ? 1 : 0;
    if (local) atomicAdd(&cnt[0], local);
    __syncthreads();
    if (cnt[0] >= K_) T = Tc;
    __syncthreads();
  }

  // ---- Stage 3: gather candidates >= T as (key desc, idx asc) composites.
  if (tid == 0) cnt[1] = 0;
  __syncthreads();
  for (int j = tid; j < N_; j += 256) {
    uint32_t k = skey[j];
    if (k >= T) {
      int p = atomicAdd(&cnt[1], 1);
      if (p < CAP)
        cand[p] = ((unsigned long long)k << 32) |
                  (unsigned long long)(0xFFFFFFFFu - (uint32_t)j);
    }
  }
  __syncthreads();
  int nc = cnt[1]; if (nc > CAP) nc = CAP;
  for (int j = tid; j < CAP; j += 256)
    if (j >= nc) cand[j] = 0ull;                  // sentinel sorts last
  __syncthreads();

  // ---- Stage 4: bitonic sort CAP entries, descending (matches lax.top_k
  // order incl. lower-index-first tie-break via ~idx in low bits).
  for (int k = 2; k <= CAP; k <<= 1) {
    for (int j = k >> 1; j > 0; j >>= 1) {
      for (int i = tid; i < CAP; i += 256) {
        int ixj = i ^ j;
        if (ixj > i) {
          unsigned long long a = cand[i], c2 = cand[ixj];
          bool swap = ((i & k) == 0) ? (a < c2) : (a > c2);
          if (swap) { cand[i] = c2; cand[ixj] = a; }
        }
      }
      __syncthreads();
    }
  }

  // ---- Stage 5: async per-lane DMA of the 200 selected boxes into LDS
  // (global_load_async_to_lds_b128: gathered 16B loads, bypasses VGPRs).
  float my_score = 0.f;
  if (tid < K_) {
    unsigned long long comp = cand[tid];
    uint32_t idx = 0xFFFFFFFFu - (uint32_t)comp;
    my_score = __uint_as_float((uint32_t)(comp >> 32));
    tsc[tid] = my_score;
    uint64_t ga = (uint64_t)(uintptr_t)(boxes + ((size_t)b * N_ + idx) * 4);
    uint32_t la = lds_base + TBOX_OFF + (uint32_t)tid * 16u;
    asm volatile("global_load_async_to_lds_b128 %0, %1, off"
                 :: "v"(la), "v"(ga) : "memory");
  }
  asm volatile("s_wait_asynccnt 0" ::: "memory");
  __syncthreads();

  // ---- Stage 6: greedy NMS. Each thread owns candidate `tid`; keep state
  // lives in a register, suppression test broadcast via __syncthreads_or.
  float ax0 = 0.f, ay0 = 0.f, ax1 = 0.f, ay1 = 0.f;
  if (tid < K_) {
    ax0 = tbox[tid * 4 + 0]; ay0 = tbox[tid * 4 + 1];
    ax1 = tbox[tid * 4 + 2]; ay1 = tbox[tid * 4 + 3];
  }
  float my_area = fmaxf(ax1 - ax0, 0.f) * fmaxf(ay1 - ay0, 0.f);
  bool my_keep = false;
  for (int i = 0; i < K_; ++i) {
    float bx0 = tbox[i * 4 + 0], by0 = tbox[i * 4 + 1];
    float bx1 = tbox[i * 4 + 2], by1 = tbox[i * 4 + 3];
    int sup = 0;
    if (tid < i && my_keep) {
      float barea = fmaxf(bx1 - bx0, 0.f) * fmaxf(by1 - by0, 0.f);
      float iw = fmaxf(fminf(ax1, bx1) - fmaxf(ax0, bx0), 0.f);
      float ih = fmaxf(fminf(ay1, by1) - fmaxf(ay0, by0), 0.f);
      float inter = iw * ih;
      float uni = my_area + barea - inter;
      float iou = inter / fmaxf(uni, 1e-9f);
      sup = (iou >= TH_IOU_) ? 1 : 0;
    }
    int s = __syncthreads_or(sup);
    if (tid == i) my_keep = (my_score >= TH_CONF_) && (s == 0);
  }

  // ---- Stage 7: masked output [B, 20, K, 5]
  if (tid < K_) {
    float m = my_keep ? 1.f : 0.f;
    size_t o = ((size_t)plane * K_ + tid) * 5;
    out[o + 0] = ax0 * m;
    out[o + 1] = ay0 * m;
    out[o + 2] = ax1 * m;
    out[o + 3] = ay1 * m;
    out[o + 4] = my_score * m;
  }
}

// ---------------------------------------------------------------------------
extern "C" void kernel_launch(void* const* d_in, const int* in_sizes, int n_in,
                              void* d_out, int out_size, void* d_ws, size_t ws_size,
                              hipStream_t stream)
{
  (void)in_sizes; (void)n_in; (void)out_size; (void)ws_size;
  const float* conf    = (const float*)d_in[0];
  const float* loc     = (const float*)d_in[1];
  const float* anchors = (const float*)d_in[2];

  // workspace: transposed scores [B,20,N] then corner boxes [B,N,4] (~76MB)
  float* scoresT = (float*)d_ws;
  float* boxes   = scoresT + (size_t)B_ * NCLS * N_;

  dim3 g1((N_ + 255) / 256, B_);
  k_softmax_decode<<<g1, 256, 0, stream>>>(conf, loc, anchors, scoresT, boxes);

  k_topk_nms<<<dim3(B_ * NCLS), 256, SMEM_SZ, stream>>>(
      scoresT, boxes, (float*)d_out);
}